// TemporalSSTGNN_78606491451782
// MI455X (gfx1250) — compile-verified
//
#include <hip/hip_runtime.h>

typedef float v2f __attribute__((ext_vector_type(2)));
typedef float v8f __attribute__((ext_vector_type(8)));

#define NODES 100000
#define EDGES 3200000
#define TBKT  8
#define FEAT  64

// ---------------------------------------------------------------------------
// Zero the segment accumulator (float4 stores, fully coalesced)
// ---------------------------------------------------------------------------
__global__ __launch_bounds__(256) void zero_f4(float4* __restrict__ p, long long n4) {
  long long i = (long long)blockIdx.x * blockDim.x + threadIdx.x;
  if (i < n4) p[i] = make_float4(0.f, 0.f, 0.f, 0.f);
}

// ---------------------------------------------------------------------------
// Segmented scatter-add: one thread per (edge, feature).
//   agg[(dst*T + t)*64 + f] += x[src*64 + f]
// Gather of x is L2-resident (25.6 MB << 192 MB L2); atomics absorbed by L2.
// ---------------------------------------------------------------------------
__global__ __launch_bounds__(256) void scatter_add(
    const float* __restrict__ x,
    const int*   __restrict__ src,
    const int*   __restrict__ dst,
    const int*   __restrict__ tix,
    float*       __restrict__ agg,
    int nedges)
{
  long long gid = (long long)blockIdx.x * blockDim.x + threadIdx.x;
  int e = (int)(gid >> 6);
  int f = (int)(gid & 63);
  if (e >= nedges) return;
  int s = src[e];
  int d = dst[e];
  int t = tix[e];
  float v = x[(size_t)s * FEAT + f];
  atomicAdd(&agg[((size_t)d * TBKT + t) * FEAT + f], v);
}

// ---------------------------------------------------------------------------
// f32 WMMA GEMM: out[nrows,64] = act(A[nrows,K] @ W[K,64] + bias)
// One wave -> one 16-row tile; 4 waves per block; 4 C accumulators (16x16 f32)
// cover the 64 output columns. K stepped by 4 via V_WMMA_F32_16X16X4_F32.
// W staged through LDS in 64x64 f32 chunks (16 KB).
//
// ISA VGPR layouts (CDNA5 ISA):
//   A 16x4 f32 : lane%16 = M, (lane/16)*2 + {0,1} = K   -> v2f per lane
//   B  4x16 f32: lane%16 = N, (lane/16)*2 + {0,1} = K   -> v2f per lane
//   C/D 16x16  : vgpr r -> M = r + (lane/16)*8, N = lane%16
// ---------------------------------------------------------------------------
template<int K, bool RELU>
__global__ __launch_bounds__(128) void gemm_wmma(
    const float* __restrict__ A,
    const float* __restrict__ W,
    const float* __restrict__ bias,
    float*       __restrict__ out,
    int nrows)
{
  __shared__ float lW[64 * FEAT];           // 16 KB chunk of W

  const int lane  = threadIdx.x & 31;
  const int wave  = threadIdx.x >> 5;
  const int row0  = (blockIdx.x * 4 + wave) * 16;
  const int mrow  = lane & 15;              // M (A) / N (B,C) within tile
  const int khalf = (lane >> 4) << 1;       // 0 or 2 : K sub-offset
  const bool active = (row0 < nrows);       // wave-uniform -> EXEC all-1s inside

  v8f acc[4] = {{}, {}, {}, {}};

  for (int k0 = 0; k0 < K; k0 += 64) {
    __syncthreads();
    // cooperative stage of W[k0:k0+64, 0:64] into LDS: 128 thr x 8 float4
    const float4* Wg = (const float4*)(W + (size_t)k0 * FEAT);
    float4*       Ls = (float4*)lW;
#pragma unroll
    for (int i = 0; i < 8; ++i)
      Ls[threadIdx.x + 128 * i] = Wg[threadIdx.x + 128 * i];
    __syncthreads();

    if (active) {
      const float* Arow = A + (size_t)(row0 + mrow) * K + k0 + khalf;
#pragma unroll
      for (int kk = 0; kk < 64; kk += 4) {
        v2f a;
        a.x = Arow[kk + 0];
        a.y = Arow[kk + 1];
        const float* b0p = &lW[(kk + khalf) * FEAT + mrow];
#pragma unroll
        for (int t = 0; t < 4; ++t) {
          v2f b;
          b.x = b0p[t * 16];                // K = kk+khalf
          b.y = b0p[t * 16 + FEAT];         // K = kk+khalf+1
          acc[t] = __builtin_amdgcn_wmma_f32_16x16x4_f32(
              /*neg_a=*/false, a, /*neg_b=*/false, b,
              /*c_mod=*/(short)0, acc[t],
              /*reuse_a=*/false, /*reuse_b=*/false);
        }
      }
    }
  }

  if (active) {
    const int rbase = row0 + ((lane >> 4) << 3);
#pragma unroll
    for (int t = 0; t < 4; ++t) {
      const int col = t * 16 + mrow;
      const float bv = bias[col];
#pragma unroll
      for (int r = 0; r < 8; ++r) {
        float v = acc[t][r] + bv;
        if (RELU) v = fmaxf(v, 0.0f);
        out[(size_t)(rbase + r) * FEAT + col] = v;
      }
    }
  }
}

// ---------------------------------------------------------------------------
// Launch: layer1 (zero, scatter, gemm+relu) -> layer2 -> ssl head
// ---------------------------------------------------------------------------
extern "C" void kernel_launch(void* const* d_in, const int* in_sizes, int n_in,
                              void* d_out, int out_size, void* d_ws, size_t ws_size,
                              hipStream_t stream)
{
  const float* x    = (const float*)d_in[0];
  const int*   eidx = (const int*)d_in[1];   // [2, E]
  const int*   tix  = (const int*)d_in[2];   // [E]
  const float* W1   = (const float*)d_in[3];
  const float* b1   = (const float*)d_in[4];
  const float* W2   = (const float*)d_in[5];
  const float* b2   = (const float*)d_in[6];
  const float* Wss  = (const float*)d_in[7];
  const float* bss  = (const float*)d_in[8];

  const int* src = eidx;
  const int* dst = eidx + EDGES;

  float* agg = (float*)d_ws;                        // [N*T, 64]  (204.8 MB)
  float* h1  = agg + (size_t)NODES * TBKT * FEAT;   // [N, 64]    (25.6 MB)
  float* h   = (float*)d_out;                       // output 0: [N, 64]
  float* ssl = h + (size_t)NODES * FEAT;            // output 1: [N, 64]

  const long long aggN = (long long)NODES * TBKT * FEAT;
  const long long n4   = aggN / 4;
  const int zblocks = (int)((n4 + 255) / 256);
  const long long sthreads = (long long)EDGES * FEAT;
  const int sblocks = (int)((sthreads + 255) / 256);
  const int gblocks = (NODES / 16 + 3) / 4;

  // ---- layer 1 ----
  zero_f4<<<zblocks, 256, 0, stream>>>((float4*)agg, n4);
  scatter_add<<<sblocks, 256, 0, stream>>>(x, src, dst, tix, agg, EDGES);
  gemm_wmma<512, true><<<gblocks, 128, 0, stream>>>(agg, W1, b1, h1, NODES);

  // ---- layer 2 ----
  zero_f4<<<zblocks, 256, 0, stream>>>((float4*)agg, n4);
  scatter_add<<<sblocks, 256, 0, stream>>>(h1, src, dst, tix, agg, EDGES);
  gemm_wmma<512, true><<<gblocks, 128, 0, stream>>>(agg, W2, b2, h, NODES);

  // ---- ssl head ----
  gemm_wmma<64, false><<<gblocks, 128, 0, stream>>>(h, Wss, bss, ssl, NODES);
}